// PrototypeClassifier_43559558316185
// MI455X (gfx1250) — compile-verified
//
#include <hip/hip_runtime.h>

typedef __attribute__((ext_vector_type(2))) float v2f;
typedef __attribute__((ext_vector_type(4))) float v4f;
typedef __attribute__((ext_vector_type(8))) float v8f;

#define DIMS      128
#define LDSTRIDE  132   // padded row stride (floats): 132 % 64 == 4 -> conflict-free b64 reads
#define TILE_M    64    // x rows per block
#define TILE_C    32    // prototypes per block

// ---------------- prep: squared norms of prototype rows and x rows ----------------
__global__ void __launch_bounds__(256)
proto_norms_kernel(const float* __restrict__ x, const float* __restrict__ p,
                   float* __restrict__ pnorm, float* __restrict__ xnorm,
                   int C, int N) {
  int wave = (blockIdx.x * blockDim.x + threadIdx.x) >> 5;
  int lane = threadIdx.x & 31;
  if (wave >= C + N) return;
  const float* row = (wave < C) ? (p + (size_t)wave * DIMS)
                                : (x + (size_t)(wave - C) * DIMS);
  v4f v = *(const v4f*)(row + lane * 4);           // 32 lanes x 4 = 128
  float s = v.x * v.x + v.y * v.y + v.z * v.z + v.w * v.w;
  #pragma unroll
  for (int off = 16; off > 0; off >>= 1)
    s += __shfl_xor(s, off, 32);
  if (lane == 0) {
    if (wave < C) pnorm[wave] = s;
    else          xnorm[wave - C] = s;
  }
}

// ---------------- main: out[m][n] = 2*(x.P^T)[m][n] - |x_m|^2 - |p_n|^2 ----------------
__global__ void __launch_bounds__(256)
proto_dist_kernel(const float* __restrict__ x, const float* __restrict__ p,
                  const float* __restrict__ pnorm, const float* __restrict__ xnorm,
                  float* __restrict__ out, int C) {
  __shared__ float lds_a[TILE_M * LDSTRIDE];   // 64*132*4  = 33792 B
  __shared__ float lds_b[TILE_C * LDSTRIDE];   // 32*132*4  = 16896 B

  const int tid  = threadIdx.x;
  const int wave = tid >> 5;
  const int lane = tid & 31;
  const int m0   = blockIdx.x * TILE_M;
  const int c0   = blockIdx.y * TILE_C;

  // Stage x block into LDS: 64x128 floats = 2048 float4, 8 per thread (coalesced).
  {
    const float* src = x + (size_t)m0 * DIMS;
    #pragma unroll
    for (int i = 0; i < 8; ++i) {
      int v    = tid + i * 256;        // 0..2047
      int row  = v >> 5;               // 0..63
      int col4 = (v & 31) * 4;         // 0..124
      v4f d = *(const v4f*)(src + row * DIMS + col4);
      *(v4f*)(&lds_a[row * LDSTRIDE + col4]) = d;
    }
  }
  // Stage prototype tile into LDS: 32x128 floats = 1024 float4, 4 per thread; zero-pad C tail.
  {
    #pragma unroll
    for (int i = 0; i < 4; ++i) {
      int v    = tid + i * 256;        // 0..1023
      int row  = v >> 5;               // 0..31
      int col4 = (v & 31) * 4;
      int c    = c0 + row;
      v4f d = {0.f, 0.f, 0.f, 0.f};
      if (c < C) d = *(const v4f*)(p + (size_t)c * DIMS + col4);
      *(v4f*)(&lds_b[row * LDSTRIDE + col4]) = d;
    }
  }
  __syncthreads();

  // 8 waves cover the 64x32 tile: wave -> (mt = w&3, ct = w>>2), each a 16x16 WMMA tile.
  const int mt   = wave & 3;
  const int ct   = wave >> 2;
  const int half = lane >> 4;          // 0: lanes 0-15 (K0,K1), 1: lanes 16-31 (K2,K3)
  const int l15  = lane & 15;

  const float* aBase = &lds_a[(mt * 16 + l15) * LDSTRIDE + half * 2]; // A[m][k], m = l15
  const float* bBase = &lds_b[(ct * 16 + l15) * LDSTRIDE + half * 2]; // B[k][n] = P[n][k], n = l15

  v8f acc = {};
  #pragma unroll
  for (int t = 0; t < 32; ++t) {       // K = 128 = 32 * 4
    v2f a = *(const v2f*)(aBase + 4 * t);
    v2f b = *(const v2f*)(bBase + 4 * t);
    acc = __builtin_amdgcn_wmma_f32_16x16x4_f32(false, a, false, b,
                                                (short)0, acc, false, false);
  }

  // Epilogue: C/D layout -> VGPR r holds M = r (+8 for upper half-wave), N = l15.
  const int n = c0 + ct * 16 + l15;
  if (n < C) {
    const float pn    = pnorm[n];
    const int   mbase = m0 + mt * 16 + half * 8;
    #pragma unroll
    for (int r = 0; r < 8; ++r) {
      int m = mbase + r;
      out[(size_t)m * C + n] = 2.0f * acc[r] - xnorm[m] - pn;
    }
  }
}

extern "C" void kernel_launch(void* const* d_in, const int* in_sizes, int n_in,
                              void* d_out, int out_size, void* d_ws, size_t ws_size,
                              hipStream_t stream) {
  const float* x = (const float*)d_in[0];   // (N, 128)
  const float* p = (const float*)d_in[1];   // (C, 128)
  float* out = (float*)d_out;

  const int N = in_sizes[0] / DIMS;         // 8192
  const int C = in_sizes[1] / DIMS;         // 1000

  float* pnorm = (float*)d_ws;              // [0 .. 1023]
  float* xnorm = pnorm + 1024;              // [1024 .. 1024+N)

  // Norm prep: one wave per row, 8 waves per block.
  int rows   = C + N;
  int nblk   = (rows + 7) / 8;
  proto_norms_kernel<<<nblk, 256, 0, stream>>>(x, p, pnorm, xnorm, C, N);

  // Main GEMM-style kernel: grid covers 8192 x 1024 (guarded to C).
  dim3 grid(N / TILE_M, (C + TILE_C - 1) / TILE_C);
  proto_dist_kernel<<<grid, 256, 0, stream>>>(x, p, pnorm, xnorm, out, C);
}